// simple_graph_conv_54820962566813
// MI455X (gfx1250) — compile-verified
//
#include <hip/hip_runtime.h>

// simple_graph_conv on MI455X (gfx1250, wave32).
//
// Factored algorithm:
//   y[b,n,o] = x[b,n,:] . W[o, 0:64]                       (WMMA f32 GEMM, 131072x64x64)
//   z[b,m,o] = x_subset[b,m,:] . (W[o,64:128]-W[o,0:64]) + bias[o]   (WMMA, 32768x64x64)
//   out[b,o,m] = max(0, max_k ( y[b, knn[b,m,k], o] + z[b,m,o] ))
//
// Uses V_WMMA_F32_16X16X4_F32 (exact f32 math, matches f32 reference numerics).

typedef float v2f __attribute__((ext_vector_type(2)));
typedef float v8f __attribute__((ext_vector_type(8)));

#define BB 4
#define NN 32768
#define MM 8192
#define KK 16
#define DD 64
#define OO 64

// MODE 0: Y = X . W[:, 0:64]^T           (rows = numTiles*16, X row stride 64)
// MODE 1: Y = X . (W[:,64:128]-W[:,0:64])^T + bias
template <int MODE>
__launch_bounds__(256)
__global__ void gemm16_wmma(const float* __restrict__ X,
                            const float* __restrict__ W,
                            const float* __restrict__ bias,
                            float* __restrict__ Y,
                            int numTiles) {
  const int lane  = threadIdx.x & 31;
  const int wib   = threadIdx.x >> 5;
  const int wpb   = blockDim.x >> 5;
  const int gwave = blockIdx.x * wpb + wib;
  const int nWav  = gridDim.x * wpb;

  const int mrow = lane & 15;        // A row / B column / C column within 16-tile
  const int grp  = lane >> 4;        // lane group selects k pair (and C row half)
  const int kb   = grp * 2;

  // Hoist B fragments (W1^T or (W2-W1)^T) for all 4 n-tiles x 16 k-steps.
  // B[k][n] = W[n*128 + k]  (n = o = 16t + mrow, k = 4s + kb + {0,1})
  v2f bf[4][16];
#pragma unroll
  for (int t = 0; t < 4; ++t) {
    const float* wrow = W + (size_t)(t * 16 + mrow) * (2 * DD);
#pragma unroll
    for (int s = 0; s < 16; ++s) {
      const int k = 4 * s + kb;
      if (MODE == 0) {
        bf[t][s] = *(const v2f*)(wrow + k);
      } else {
        v2f p = *(const v2f*)(wrow + DD + k);   // W2 slice
        v2f q = *(const v2f*)(wrow + k);        // W1 slice
        bf[t][s] = p - q;
      }
    }
  }

  float bv[4];
#pragma unroll
  for (int t = 0; t < 4; ++t)
    bv[t] = (MODE == 1) ? bias[t * 16 + mrow] : 0.0f;

  for (int tile = gwave; tile < numTiles; tile += nWav) {
    const float* xrow = X + (size_t)(tile * 16 + mrow) * DD + kb;

    v8f acc[4];
#pragma unroll
    for (int t = 0; t < 4; ++t)
#pragma unroll
      for (int i = 0; i < 8; ++i) acc[t][i] = bv[t];

#pragma unroll
    for (int s = 0; s < 16; ++s) {
      // A fragment: A[mrow][4s + kb + {0,1}], contiguous float2
      v2f a = *(const v2f*)(xrow + 4 * s);
#pragma unroll
      for (int t = 0; t < 4; ++t) {
        acc[t] = __builtin_amdgcn_wmma_f32_16x16x4_f32(
            /*neg_a=*/false, a, /*neg_b=*/false, bf[t][s],
            /*c_mod=*/(short)0, acc[t], /*reuse_a=*/false, /*reuse_b=*/false);
      }
    }

    // C layout: VGPR r, lanes 0-15 -> row r, lanes 16-31 -> row r+8; col = mrow.
    float* yout = Y + (size_t)(tile * 16 + 8 * grp) * OO;
#pragma unroll
    for (int t = 0; t < 4; ++t)
#pragma unroll
      for (int r = 0; r < 8; ++r)
        yout[(size_t)r * OO + t * 16 + mrow] = acc[t][r];
  }
}

// One wave per (b, m): lane covers two output channels o = {2*lane, 2*lane+1}.
__launch_bounds__(256)
__global__ void gather_max_kernel(const float* __restrict__ y,
                                  const float* __restrict__ z,
                                  const int* __restrict__ knn,
                                  float* __restrict__ out) {
  const int lane  = threadIdx.x & 31;
  const int gwave = (blockIdx.x * blockDim.x + threadIdx.x) >> 5;  // in [0, B*M)
  const int b = gwave >> 13;       // / MM
  const int m = gwave & (MM - 1);
  const int o0 = lane * 2;

  const v2f zv = *(const v2f*)(z + (size_t)gwave * OO + o0);
  v2f acc;
  acc.x = 0.0f;   // max_k relu(v) == max(0, max_k v)
  acc.y = 0.0f;

  const int* kn = knn + (size_t)gwave * KK;
#pragma unroll
  for (int k = 0; k < KK; ++k) {
    const int idx = kn[k];
    const v2f yv = *(const v2f*)(y + ((size_t)b * NN + idx) * OO + o0);
    v2f s = yv + zv;
    acc.x = fmaxf(acc.x, s.x);
    acc.y = fmaxf(acc.y, s.y);
  }

  float* op = out + ((size_t)b * OO + o0) * MM + m;
  op[0]  = acc.x;
  op[MM] = acc.y;
}

extern "C" void kernel_launch(void* const* d_in, const int* in_sizes, int n_in,
                              void* d_out, int out_size, void* d_ws, size_t ws_size,
                              hipStream_t stream) {
  const float* x    = (const float*)d_in[0];   // (B, N, D)
  const float* xs   = (const float*)d_in[1];   // (B, M, D)
  const int*   knn  = (const int*)d_in[2];     // (B, M, K)
  const float* W    = (const float*)d_in[3];   // (O, 2D)
  const float* bias = (const float*)d_in[4];   // (O,)
  float* out = (float*)d_out;                  // (B, O, M)

  float* y = (float*)d_ws;                         // B*N*O floats = 33.5 MB
  float* z = y + (size_t)BB * NN * OO;             // B*M*O floats =  8.4 MB

  // y = x . W1^T   (131072 rows -> 8192 tiles; 2048 waves, 4 tiles each)
  gemm16_wmma<0><<<256, 256, 0, stream>>>(x, W, bias, y, (BB * NN) / 16);
  // z = x_subset . (W2-W1)^T + bias  (32768 rows -> 2048 tiles)
  gemm16_wmma<1><<<256, 256, 0, stream>>>(xs, W, bias, z, (BB * MM) / 16);
  // out = max(0, max_k (gather(y) + z))  : one wave per (b,m)
  gather_max_kernel<<<(BB * MM * 32) / 256, 256, 0, stream>>>(y, z, knn, out);
}